// GINENet_3118146257466
// MI455X (gfx1250) — compile-verified
//
#include <hip/hip_runtime.h>
#include <hip/hip_bf16.h>

// GINE (GINEConv x3 + mean-pool + MLP head) for MI455X / gfx1250.
// Memory-bound (~3 GB traffic vs ~20 GFLOP) -> FP32 everywhere, WMMA
// V_WMMA_F32_16X16X4_F32 for all dense GEMMs, repacked weight layout so each
// B-chunk is one coalesced b64 load, async global->LDS staging of the edge
// weights, global_atomic_add_f32 scatters for message aggregation.

typedef __attribute__((ext_vector_type(2))) float v2f;
typedef __attribute__((ext_vector_type(4))) float v4f;
typedef __attribute__((ext_vector_type(8))) float v8f;
typedef __attribute__((ext_vector_type(4))) int   v4i;

#define N_NODES 50000
#define N_EDGES 800000
#define DIM     128
#define EDIM    16
#define N_LAYER 3
#define N_GRAPH 64

// Repacked weight sizes (floats): kchunks * 8 ntiles * 32 lanes * 2
#define WP_NODE (32 * 8 * 32 * 2)   // 16384 floats = 64 KB  (K=128)
#define WP_EDGE (4 * 8 * 32 * 2)    // 2048 floats  = 8 KB   (K=16)

__device__ __forceinline__ v8f wmma_f32_16x16x4(v2f a, v2f b, v8f c) {
  // 8 args: (neg_a, A, neg_b, B, c_mod, C, reuse_a, reuse_b)
  return __builtin_amdgcn_wmma_f32_16x16x4_f32(false, a, false, b, (short)0, c,
                                               false, false);
}

__device__ __forceinline__ float frelu(float v) { return v > 0.f ? v : 0.f; }

// Async global->LDS 16-byte copy (ASYNCcnt path) with safe fallback.
// Builtin signature (per compiler diagnostic): pointers to int4 vectors.
__device__ __forceinline__ void async_copy_b128(const float* gsrc, float* ldst) {
#if __has_builtin(__builtin_amdgcn_global_load_async_to_lds_b128)
  __builtin_amdgcn_global_load_async_to_lds_b128((v4i*)gsrc, (v4i*)ldst, 0, 0);
#else
  *(v4f*)ldst = *(const v4f*)gsrc;
#endif
}

__device__ __forceinline__ void wait_async_all() {
#if __has_builtin(__builtin_amdgcn_s_wait_asynccnt)
  __builtin_amdgcn_s_wait_asynccnt(0);
#else
  asm volatile("s_wait_asynccnt 0x0" ::: "memory");
#endif
}

// -------------------------------------------------------------------------
// Zero-fill scratch
__global__ void gine_zero_kernel(float* __restrict__ p, int n) {
  int i = blockIdx.x * blockDim.x + threadIdx.x;
  if (i < n) p[i] = 0.f;
}

// -------------------------------------------------------------------------
// Repack W[K][128] (row-major) into B-chunk layout:
//   Wp[((kc*8 + nt)*32 + lane)*2 + j] = W[(kc*4 + 2*(lane>>4) + j)*128
//                                         + nt*16 + (lane&15)]
// so a wave's B operand for (kc, nt) is one lane-linear float2 (b64) load.
__global__ void gine_repack_kernel(const float* __restrict__ W,
                                   float* __restrict__ Wp, int kchunks) {
  int idx = blockIdx.x * blockDim.x + threadIdx.x;   // over kchunks*8*32
  if (idx >= kchunks * 8 * 32) return;
  int lane = idx & 31;
  int nt = (idx >> 5) & 7;
  int kc = idx >> 8;
  int col = nt * 16 + (lane & 15);
  int krow = kc * 4 + 2 * (lane >> 4);
  Wp[(size_t)idx * 2 + 0] = W[(size_t)krow * DIM + col];
  Wp[(size_t)idx * 2 + 1] = W[(size_t)(krow + 1) * DIM + col];
}

// -------------------------------------------------------------------------
// Fused edge stage: e = edge_attr @ We + be ; m = relu(h[src] + e) ;
// agg[dst] += m.  One wave per 16-edge tile; We/be staged in LDS via
// async global->LDS so the VMEM pipe is free for gathers + atomics.
__global__ void gine_edge_kernel(const float* __restrict__ h,     // [N,128]
                                 const float* __restrict__ ea,    // [E,16]
                                 const int* __restrict__ src,
                                 const int* __restrict__ dst,
                                 const float* __restrict__ Wep,   // [WP_EDGE]
                                 const float* __restrict__ be,    // [128]
                                 float* __restrict__ agg,         // [N,128]
                                 int nEdges) {
  __shared__ float sW[WP_EDGE];   // 8 KB
  __shared__ float sB[DIM];       // 512 B
  const int tid = threadIdx.x;    // blockDim.x == 256

  // Stage repacked We (512 x b128) + bias (32 x b128) asynchronously.
#pragma unroll
  for (int i = 0; i < 2; ++i) {
    int idx = i * 256 + tid;
    async_copy_b128(Wep + (size_t)idx * 4, &sW[idx * 4]);
  }
  if (tid < 32) async_copy_b128(be + (size_t)tid * 4, &sB[tid * 4]);
  wait_async_all();
  __syncthreads();

  const int lane  = tid & 31;
  const int wave  = blockIdx.x * (blockDim.x >> 5) + (tid >> 5);
  const int base  = wave * 16;
  if (base >= nEdges) return;              // wave-uniform: EXEC all-ones
  const int rlo   = lane & 15;
  const int khalf = lane >> 4;

  // A-matrix chunks (16x4 f32): lane (row=rlo, K-half=khalf) -> float2 load.
  v2f a[4];
#pragma unroll
  for (int kc = 0; kc < 4; ++kc) {
    const float* p = ea + (size_t)(base + rlo) * EDIM + kc * 4 + khalf * 2;
    a[kc] = *(const v2f*)p;
  }

  // 8 column tiles of 16, K = 16 in 4 chunks; B from LDS (ds_load_b64).
  v8f c[8] = {};
#pragma unroll
  for (int nt = 0; nt < 8; ++nt) {
#pragma unroll
    for (int kc = 0; kc < 4; ++kc) {
      v2f b = *(const v2f*)(&sW[(size_t)((kc * 8 + nt) * 32 + lane) * 2]);
      c[nt] = wmma_f32_16x16x4(a[kc], b, c[nt]);
    }
  }

  float bias[8];
#pragma unroll
  for (int nt = 0; nt < 8; ++nt) bias[nt] = sB[nt * 16 + rlo];

  // Epilogue: C VGPR r holds (edge row = r + 8*khalf, col = nt*16 + rlo).
#pragma unroll
  for (int r = 0; r < 8; ++r) {
    const int e = base + r + 8 * khalf;
    const int s = src[e];
    const int d = dst[e];
    const float* hrow = h + (size_t)s * DIM;
    float* arow = agg + (size_t)d * DIM;
#pragma unroll
    for (int nt = 0; nt < 8; ++nt) {
      const int col = nt * 16 + rlo;
      float v = frelu(c[nt][r] + bias[nt] + hrow[col]);
      atomicAdd(arow + col, v);            // global_atomic_add_f32
    }
  }
}

// -------------------------------------------------------------------------
// Node MLP: z = h + agg ; hout = relu(relu(z@W1+b1)@W2+b2).
// One wave per 16-node tile; intermediate re-laid out via per-wave LDS;
// weights read from repacked layout as coalesced b64 loads (L2/WGP$-hot).
#define NMLP_WAVES 4
__global__ void gine_node_mlp_kernel(const float* __restrict__ h,
                                     const float* __restrict__ agg,
                                     const float* __restrict__ W1p,  // [WP_NODE]
                                     const float* __restrict__ b1,
                                     const float* __restrict__ W2p,  // [WP_NODE]
                                     const float* __restrict__ b2,
                                     float* __restrict__ hout,
                                     int nNodes) {
  __shared__ float tbuf[NMLP_WAVES][16 * DIM];   // 32 KB / block
  const int lane  = threadIdx.x & 31;
  const int wslot = threadIdx.x >> 5;
  const int wave  = blockIdx.x * NMLP_WAVES + wslot;
  const int base  = wave * 16;
  if (base >= nNodes) return;              // wave-uniform
  const int rlo   = lane & 15;
  const int khalf = lane >> 4;

  if (lane == 0) { __builtin_prefetch(W1p, 0, 0); __builtin_prefetch(W2p, 0, 0); }

  // ---- GEMM1: (h + agg) @ W1 ----
  v8f c[8] = {};
  for (int kc = 0; kc < 32; ++kc) {        // K = 128 in chunks of 4
    const size_t off = (size_t)(base + rlo) * DIM + kc * 4 + khalf * 2;
    v2f hz = *(const v2f*)(h + off);
    v2f az = *(const v2f*)(agg + off);
    v2f a;
    a.x = hz.x + az.x;
    a.y = hz.y + az.y;
#pragma unroll
    for (int nt = 0; nt < 8; ++nt) {
      v2f b = *(const v2f*)(W1p + (size_t)((kc * 8 + nt) * 32 + lane) * 2);
      c[nt] = wmma_f32_16x16x4(a, b, c[nt]);
    }
  }

  // t = relu(c + b1) -> LDS in [row][col] layout (WMMA D -> A re-layout).
#pragma unroll
  for (int nt = 0; nt < 8; ++nt) {
    const int col = nt * 16 + rlo;
    const float bb = b1[col];
#pragma unroll
    for (int r = 0; r < 8; ++r) {
      const int row = r + 8 * khalf;
      tbuf[wslot][row * DIM + col] = frelu(c[nt][r] + bb);
    }
  }
  asm volatile("s_wait_dscnt 0x0" ::: "memory");   // order ds_store -> ds_load

  // ---- GEMM2: t @ W2 ----
  v8f c2[8] = {};
  for (int kc = 0; kc < 32; ++kc) {
    v2f a = *(const v2f*)(&tbuf[wslot][rlo * DIM + kc * 4 + khalf * 2]);
#pragma unroll
    for (int nt = 0; nt < 8; ++nt) {
      v2f b = *(const v2f*)(W2p + (size_t)((kc * 8 + nt) * 32 + lane) * 2);
      c2[nt] = wmma_f32_16x16x4(a, b, c2[nt]);
    }
  }

  // hout = relu(c2 + b2)
#pragma unroll
  for (int nt = 0; nt < 8; ++nt) {
    const int col = nt * 16 + rlo;
    const float bb = b2[col];
#pragma unroll
    for (int r = 0; r < 8; ++r) {
      const int row = r + 8 * khalf;
      hout[(size_t)(base + row) * DIM + col] = frelu(c2[nt][r] + bb);
    }
  }
}

// -------------------------------------------------------------------------
// Global mean pool: gsum[g] += h[n], gcnt[g] += 1.
__global__ void gine_pool_kernel(const float* __restrict__ h,
                                 const int* __restrict__ batch,
                                 float* __restrict__ gsum,
                                 float* __restrict__ gcnt, int nNodes) {
  int idx = blockIdx.x * blockDim.x + threadIdx.x;   // over nNodes*128
  if (idx >= nNodes * DIM) return;
  int n = idx >> 7, col = idx & (DIM - 1);
  int g = batch[n];
  atomicAdd(&gsum[(size_t)g * DIM + col], h[idx]);
  if (col == 0) atomicAdd(&gcnt[g], 1.0f);
}

// -------------------------------------------------------------------------
// Head: out[g] = relu(mean_g @ Wh1 + bh1) @ Wh2 + bh2. One block per graph.
__global__ void gine_head_kernel(const float* __restrict__ gsum,
                                 const float* __restrict__ gcnt,
                                 const float* __restrict__ Wh1,  // [128,128]
                                 const float* __restrict__ bh1,
                                 const float* __restrict__ Wh2,  // [128,1]
                                 const float* __restrict__ bh2,
                                 float* __restrict__ out) {
  __shared__ float red[DIM];
  const int g = blockIdx.x, t = threadIdx.x;
  float cnt = gcnt[g];
  cnt = cnt > 1.f ? cnt : 1.f;
  const float inv = 1.f / cnt;
  float acc = bh1[t];
  for (int k = 0; k < DIM; ++k)
    acc += gsum[(size_t)g * DIM + k] * inv * Wh1[(size_t)k * DIM + t];
  red[t] = frelu(acc) * Wh2[t];
  __syncthreads();
  for (int s = DIM / 2; s > 0; s >>= 1) {
    if (t < s) red[t] += red[t + s];
    __syncthreads();
  }
  if (t == 0) out[g] = red[0] + bh2[0];
}

// -------------------------------------------------------------------------
extern "C" void kernel_launch(void* const* d_in, const int* in_sizes, int n_in,
                              void* d_out, int out_size, void* d_ws,
                              size_t ws_size, hipStream_t stream) {
  (void)in_sizes; (void)n_in; (void)out_size; (void)ws_size;
  const float* x     = (const float*)d_in[0];
  const int*   ei    = (const int*)d_in[1];
  const int*   batch = (const int*)d_in[2];
  const float* ea    = (const float*)d_in[3];
  const float* W1    = (const float*)d_in[4];
  const float* b1    = (const float*)d_in[5];
  const float* W2    = (const float*)d_in[6];
  const float* b2    = (const float*)d_in[7];
  const float* We    = (const float*)d_in[8];
  const float* be    = (const float*)d_in[9];
  const float* Wh1   = (const float*)d_in[10];
  const float* bh1   = (const float*)d_in[11];
  const float* Wh2   = (const float*)d_in[12];
  const float* bh2   = (const float*)d_in[13];

  const int* src = ei;              // edge_index[0]
  const int* dst = ei + N_EDGES;    // edge_index[1]

  float* ws    = (float*)d_ws;
  float* hbuf0 = ws;                               // N*128
  float* hbuf1 = hbuf0 + (size_t)N_NODES * DIM;    // N*128
  float* agg   = hbuf1 + (size_t)N_NODES * DIM;    // N*128
  float* gsum  = agg + (size_t)N_NODES * DIM;      // 64*128
  float* gcnt  = gsum + (size_t)N_GRAPH * DIM;     // 64
  float* W1p   = gcnt + N_GRAPH;                   // 3 * WP_NODE (16B aligned)
  float* W2p   = W1p + (size_t)N_LAYER * WP_NODE;  // 3 * WP_NODE
  float* Wep   = W2p + (size_t)N_LAYER * WP_NODE;  // 3 * WP_EDGE

  // One-time weight repacks (tiny).
  for (int l = 0; l < N_LAYER; ++l) {
    gine_repack_kernel<<<(32 * 8 * 32 + 255) / 256, 256, 0, stream>>>(
        W1 + (size_t)l * DIM * DIM, W1p + (size_t)l * WP_NODE, 32);
    gine_repack_kernel<<<(32 * 8 * 32 + 255) / 256, 256, 0, stream>>>(
        W2 + (size_t)l * DIM * DIM, W2p + (size_t)l * WP_NODE, 32);
    gine_repack_kernel<<<(4 * 8 * 32 + 255) / 256, 256, 0, stream>>>(
        We + (size_t)l * EDIM * DIM, Wep + (size_t)l * WP_EDGE, 4);
  }

  const float* hcur = x;
  float* houts[N_LAYER] = {hbuf0, hbuf1, hbuf0};

  const int edgeWavesPerBlock = 8;                      // 256 threads
  const int edgeTiles = N_EDGES / 16;                   // 50000, exact
  const int nodeTiles = N_NODES / 16;                   // 3125, exact

  for (int l = 0; l < N_LAYER; ++l) {
    gine_zero_kernel<<<(N_NODES * DIM + 255) / 256, 256, 0, stream>>>(
        agg, N_NODES * DIM);
    gine_edge_kernel<<<(edgeTiles + edgeWavesPerBlock - 1) / edgeWavesPerBlock,
                       edgeWavesPerBlock * 32, 0, stream>>>(
        hcur, ea, src, dst, Wep + (size_t)l * WP_EDGE, be + (size_t)l * DIM,
        agg, N_EDGES);
    gine_node_mlp_kernel<<<(nodeTiles + NMLP_WAVES - 1) / NMLP_WAVES,
                           NMLP_WAVES * 32, 0, stream>>>(
        hcur, agg, W1p + (size_t)l * WP_NODE, b1 + (size_t)l * DIM,
        W2p + (size_t)l * WP_NODE, b2 + (size_t)l * DIM, houts[l], N_NODES);
    hcur = houts[l];
  }

  gine_zero_kernel<<<(N_GRAPH * DIM + N_GRAPH + 255) / 256, 256, 0, stream>>>(
      gsum, N_GRAPH * DIM + N_GRAPH);
  gine_pool_kernel<<<(N_NODES * DIM + 255) / 256, 256, 0, stream>>>(
      hcur, batch, gsum, gcnt, N_NODES);
  gine_head_kernel<<<N_GRAPH, DIM, 0, stream>>>(gsum, gcnt, Wh1, bh1, Wh2, bh2,
                                                (float*)d_out);
}